// HGNN_68143951119110
// MI455X (gfx1250) — compile-verified
//
#include <hip/hip_runtime.h>
#include <hip/hip_bf16.h>

typedef __attribute__((ext_vector_type(2))) float v2f;
typedef __attribute__((ext_vector_type(8))) float v8f;

#if __has_builtin(__builtin_amdgcn_global_load_async_to_lds_b128)
#define HAVE_ASYNC_LDS 1
#else
#define HAVE_ASYNC_LDS 0
#endif

// The async-to-LDS builtin expects (int4_gvec __device__* src, int4_gvec __shared__* dst,
// imm offset, imm cpol) where int4_gvec is a GCC-style vector_size(16) int.
typedef int v4i_vs __attribute__((vector_size(16)));
#define AS1_V4I(p) ((__attribute__((address_space(1))) v4i_vs*)(p))
#define AS3_V4I(p) ((__attribute__((address_space(3))) v4i_vs*)(p))

template <int N>
__device__ __forceinline__ void wait_asynccnt() {
#if __has_builtin(__builtin_amdgcn_s_wait_asynccnt)
    __builtin_amdgcn_s_wait_asynccnt((unsigned short)N);
#else
    asm volatile("s_wait_asynccnt %0" :: "n"(N) : "memory");
#endif
}

// ---------------------------------------------------------------------------
// Utility kernels
// ---------------------------------------------------------------------------

__global__ void zerof_kernel(float* p, long n) {
    long i = (long)blockIdx.x * blockDim.x + threadIdx.x;
    long st = (long)gridDim.x * blockDim.x;
    for (; i < n; i += st) p[i] = 0.f;
}

// d[n] = rsqrt(sum_e H[n,e]) ; one block per row, coalesced in e
__global__ void rowsum_rsqrt_kernel(const float* __restrict__ H, float* __restrict__ d,
                                    int Nn, int Ee) {
    __shared__ float red[256];
    const int n = blockIdx.x;
    float s = 0.f;
    for (int e = threadIdx.x; e < Ee; e += 256) s += H[(size_t)n * Ee + e];
    red[threadIdx.x] = s;
    __syncthreads();
    for (int off = 128; off > 0; off >>= 1) {
        if (threadIdx.x < off) red[threadIdx.x] += red[threadIdx.x + off];
        __syncthreads();
    }
    if (threadIdx.x == 0) d[n] = rsqrtf(red[0]);
}

// acc[e] += sum over a row-chunk of H[:,e] ; coalesced across e, split over rows
__global__ void colsum_partial_kernel(const float* __restrict__ H, float* __restrict__ acc,
                                      int Nn, int Ee, int chunk) {
    const int e = blockIdx.x * blockDim.x + threadIdx.x;
    if (e >= Ee) return;
    int r0 = blockIdx.y * chunk;
    int r1 = r0 + chunk; if (r1 > Nn) r1 = Nn;
    float s = 0.f;
    for (int r = r0; r < r1; r++) s += H[(size_t)r * Ee + e];
    atomicAdd(&acc[e], s);
}

__global__ void recip_inplace_kernel(float* __restrict__ x, int n) {
    const int i = blockIdx.x * blockDim.x + threadIdx.x;
    if (i < n) x[i] = 1.f / x[i];
}

// out[c][r] = (scale ? scale[c] : 1) * in[r][c]   (in is R x C, row-major)
__global__ void transpose_scaled_kernel(const float* __restrict__ in, float* __restrict__ out,
                                        int R, int C, const float* __restrict__ scale) {
    __shared__ float t[32][33];
    const int c0 = blockIdx.x * 32, r0 = blockIdx.y * 32;
    const int tx = threadIdx.x, ty = threadIdx.y;   // (32, 8)
    #pragma unroll
    for (int i = 0; i < 32; i += 8) {
        int r = r0 + ty + i, c = c0 + tx;
        t[ty + i][tx] = (r < R && c < C) ? in[(size_t)r * C + c] : 0.f;
    }
    __syncthreads();
    #pragma unroll
    for (int i = 0; i < 32; i += 8) {
        int c = c0 + ty + i, r = r0 + tx;
        if (c < C && r < R) {
            float v = t[tx][ty + i];
            if (scale) v *= scale[c];
            out[(size_t)c * R + r] = v;
        }
    }
}

// BatchNorm statistics: per-feature partial sums / sums of squares
__global__ void bn_stats_kernel(const float* __restrict__ X, float* __restrict__ sacc,
                                float* __restrict__ sqacc, int Nn, int chunk) {
    const int j = threadIdx.x;          // feature; blockDim.x == DH
    const int DH = blockDim.x;
    int r0 = blockIdx.x * chunk;
    int r1 = r0 + chunk; if (r1 > Nn) r1 = Nn;
    float s = 0.f, q = 0.f;
    for (int r = r0; r < r1; r++) {
        float v = X[(size_t)r * DH + j];
        s += v; q += v * v;
    }
    atomicAdd(&sacc[j], s);
    atomicAdd(&sqacc[j], q);
}

__global__ void bn_finalize_kernel(const float* __restrict__ sacc, const float* __restrict__ sqacc,
                                   const float* __restrict__ gamma, const float* __restrict__ beta,
                                   float* __restrict__ bnA, float* __restrict__ bnB, int Nn) {
    const int j = threadIdx.x;
    float mean = sacc[j] / (float)Nn;
    float var  = sqacc[j] / (float)Nn - mean * mean;
    float sc   = gamma[j] * rsqrtf(var + 1e-5f);
    bnA[j] = sc;
    bnB[j] = beta[j] - mean * sc;
}

// x = LeakyReLU_0.1(x * bnA[j] + bnB[j]) in place
__global__ void bn_apply_lrelu_kernel(float* __restrict__ X, const float* __restrict__ bnA,
                                      const float* __restrict__ bnB, long total, int DH) {
    long i = (long)blockIdx.x * blockDim.x + threadIdx.x;
    long st = (long)gridDim.x * blockDim.x;
    for (; i < total; i += st) {
        int j = (int)(i % DH);
        float v = X[i] * bnA[j] + bnB[j];
        X[i] = (v >= 0.f) ? v : 0.1f * v;
    }
}

// ---------------------------------------------------------------------------
// Unified fp32 WMMA GEMM:  C[M][Nc] (+)= rowScale ⊙ (A @ B + colBias)
//   A_MK=true : A stored [M][K] row-major; LDS tile kept as [m][k] so each
//               A fragment is one aligned ds_load_b64.
//   A_MK=false: A stored [K][M] row-major; LDS tile [k][m].
//   B stored [K][Nc] row-major; LDS tile [k][n].
//   Split-K via gridDim.z (kChunk rows per slice) with fp32 atomic epilogue.
//   Interior tiles use double-buffered GLOBAL_LOAD_ASYNC_TO_LDS_B128 copies
//   (ASYNCcnt-tracked) to overlap HBM streaming of H with WMMA compute.
// ---------------------------------------------------------------------------

template <bool A_MK>
__global__ __launch_bounds__(256)
void gemm_f32_wmma_kernel(const float* __restrict__ A, int lda,
                          const float* __restrict__ B, int ldb,
                          float* __restrict__ C, int ldc,
                          int M, int Nc, int K, int kChunk,
                          const float* __restrict__ colBias,
                          const float* __restrict__ rowScale,
                          int atomicOut) {
    constexpr int BM = 128, BN = 128, KC = 24, THREADS = 256, NT = BN / 16;
    constexpr int AD0 = A_MK ? BM : KC;          // LDS A tile shape
    constexpr int AD1 = A_MK ? (KC + 4) : (BM + 4);
    // rows padded by 4 floats: keeps every row 16B-aligned for async B128
    // destinations while staying bank-conflict-free for fragment reads.
    __shared__ __align__(16) float a_s[2][AD0][AD1];
    __shared__ __align__(16) float b_s[2][KC][BN + 4];

    const int tid  = threadIdx.x;
    const int wave = tid >> 5;
    const int lane = tid & 31;
    const int l16  = lane & 15;
    const int half = lane >> 4;           // K-half select for A/B fragments
    const int n0   = blockIdx.x * BN;
    const int m0   = blockIdx.y * BM;
    const int kBeg = blockIdx.z * kChunk;
    int kEnd = kBeg + kChunk; if (kEnd > K) kEnd = K;

    v8f acc[NT];
    const v8f vzero = {0.f, 0.f, 0.f, 0.f, 0.f, 0.f, 0.f, 0.f};
    #pragma unroll
    for (int i = 0; i < NT; i++) acc[i] = vzero;

    // ---- predicated (exact) tile loader: used for all ragged tiles --------
    auto loadTileManual = [&](int buf, int kt) {
        if constexpr (A_MK) {
            for (int i = tid; i < KC * BM; i += THREADS) {
                int k = i % KC, m = i / KC;
                int gm = m0 + m, gk = kt + k;
                a_s[buf][m][k] = (gm < M && gk < kEnd) ? A[(size_t)gm * lda + gk] : 0.f;
            }
        } else {
            for (int i = tid; i < KC * BM; i += THREADS) {
                int m = i & (BM - 1), k = i >> 7;
                int gm = m0 + m, gk = kt + k;
                a_s[buf][k][m] = (gm < M && gk < kEnd) ? A[(size_t)gk * lda + gm] : 0.f;
            }
        }
        for (int i = tid; i < KC * BN; i += THREADS) {
            int n = i & (BN - 1), k = i >> 7;
            int gn = n0 + n, gk = kt + k;
            b_s[buf][k][n] = (gn < Nc && gk < kEnd) ? B[(size_t)gk * ldb + gn] : 0.f;
            if (gk + KC < kEnd && gn < Nc)
                __builtin_prefetch(&B[(size_t)(gk + KC) * ldb + gn], 0, 1);
        }
    };

#if HAVE_ASYNC_LDS
    // ---- async tile loader: interior tiles only (no predication needed) ---
    // 3 A-chunks + 3 B-chunks of 16B per thread -> 6 async ops per wave/tile.
    auto issueTileAsync = [&](int buf, int kt) {
        if constexpr (A_MK) {
            #pragma unroll
            for (int i = 0; i < 3; i++) {
                int c = tid + i * THREADS;            // 768 chunks of 4 floats
                int m = c / 6, kq = (c % 6) * 4;
                __builtin_amdgcn_global_load_async_to_lds_b128(
                    AS1_V4I(&A[(size_t)(m0 + m) * lda + kt + kq]),
                    AS3_V4I(&a_s[buf][m][kq]), 0, 0);
            }
        } else {
            #pragma unroll
            for (int i = 0; i < 3; i++) {
                int c = tid + i * THREADS;
                int k = c >> 5, mq = (c & 31) * 4;
                __builtin_amdgcn_global_load_async_to_lds_b128(
                    AS1_V4I(&A[(size_t)(kt + k) * lda + m0 + mq]),
                    AS3_V4I(&a_s[buf][k][mq]), 0, 0);
            }
        }
        #pragma unroll
        for (int i = 0; i < 3; i++) {
            int c = tid + i * THREADS;
            int k = c >> 5, nq = (c & 31) * 4;
            __builtin_amdgcn_global_load_async_to_lds_b128(
                AS1_V4I(&B[(size_t)(kt + k) * ldb + n0 + nq]),
                AS3_V4I(&b_s[buf][k][nq]), 0, 0);
        }
    };
#endif

    // ---- WMMA over one staged tile ---------------------------------------
    auto computeTile = [&](int buf) {
        #pragma unroll
        for (int kk = 0; kk < KC; kk += 4) {
            const int ka = kk + half * 2;  // lanes 0-15: K=+0/+1, lanes 16-31: K=+2/+3
            v2f a;
            if constexpr (A_MK) {
                a = *(const v2f*)&a_s[buf][wave * 16 + l16][ka];   // one ds_load_b64
            } else {
                a.x = a_s[buf][ka][wave * 16 + l16];
                a.y = a_s[buf][ka + 1][wave * 16 + l16];
            }
            #pragma unroll
            for (int nt = 0; nt < NT; nt++) {
                v2f b;
                b.x = b_s[buf][ka][nt * 16 + l16];
                b.y = b_s[buf][ka + 1][nt * 16 + l16];
                acc[nt] = __builtin_amdgcn_wmma_f32_16x16x4_f32(
                    false, a, false, b, (short)0, acc[nt], false, false);
            }
        }
    };

#if HAVE_ASYNC_LDS
    const bool interiorNM = (n0 + BN <= Nc) && (m0 + BM <= M);
    if (interiorNM) {
        const int nT = (kEnd - kBeg + KC - 1) / KC;
        auto kAt    = [&](int i) { return kBeg + i * KC; };
        auto isFull = [&](int i) { return kAt(i) + KC <= kEnd; };
        if (nT > 0) {
            if (isFull(0)) issueTileAsync(0, kAt(0));
            else           loadTileManual(0, kAt(0));
        }
        int parity = 0;
        for (int i = 0; i < nT; i++) {
            const bool curFull  = isFull(i);
            const bool haveNext = (i + 1 < nT);
            const bool nextFull = haveNext && isFull(i + 1);
            if (haveNext) {
                if (nextFull) issueTileAsync(parity ^ 1, kAt(i + 1));
                else          loadTileManual(parity ^ 1, kAt(i + 1));
            }
            if (curFull) {
                if (nextFull) wait_asynccnt<6>();  // drain current tile's copies
                else          wait_asynccnt<0>();
            }
            __syncthreads();
            computeTile(parity);
            __syncthreads();
            parity ^= 1;
        }
    } else
#endif
    {
        // ragged blocks: fully predicated single-buffered loop
        for (int kt = kBeg; kt < kEnd; kt += KC) {
            loadTileManual(0, kt);
            __syncthreads();
            computeTile(0);
            __syncthreads();
        }
    }

    // ---- epilogue: C/D layout -> VGPR g holds row (half*8+g), lane n = l16 ----
    #pragma unroll
    for (int nt = 0; nt < NT; nt++) {
        int gn = n0 + nt * 16 + l16;
        if (gn >= Nc) continue;
        float cb = colBias ? colBias[gn] : 0.f;
        #pragma unroll
        for (int g = 0; g < 8; g++) {
            int gm = m0 + wave * 16 + half * 8 + g;
            if (gm >= M) continue;
            float v = acc[nt][g];
            if (atomicOut) {
                if (rowScale) v *= rowScale[gm];   // row scale commutes with split-K sum
                atomicAdd(&C[(size_t)gm * ldc + gn], v);
            } else {
                v += cb;
                if (rowScale) v *= rowScale[gm];
                C[(size_t)gm * ldc + gn] = v;
            }
        }
    }
}

// ---------------------------------------------------------------------------
// Host orchestration
// ---------------------------------------------------------------------------

extern "C" void kernel_launch(void* const* d_in, const int* in_sizes, int n_in,
                              void* d_out, int out_size, void* d_ws, size_t ws_size,
                              hipStream_t stream) {
    const float* X     = (const float*)d_in[0];
    const float* H     = (const float*)d_in[1];
    const float* W1    = (const float*)d_in[2];
    const float* b1    = (const float*)d_in[3];
    const float* gamma = (const float*)d_in[4];
    const float* beta  = (const float*)d_in[5];
    const float* W2    = (const float*)d_in[6];
    const float* b2    = (const float*)d_in[7];
    float* out = (float*)d_out;

    const int DH = in_sizes[3];              // 128
    const int NC = in_sizes[7];              // 80
    const int NB = in_sizes[2] / DH;         // 64
    const int Nn = in_sizes[0] / NB;         // 10000
    const int Ee = in_sizes[1] / Nn;         // 5000

    float* ws = (float*)d_ws;
    size_t ofs = 0;
    auto wsa = [&](size_t n) { float* p = ws + ofs; ofs += (n + 63) & ~(size_t)63; return p; };
    float* dvec  = wsa(Nn);                  // d = Dv^-1/2
    float* invDe = wsa(Ee);                  // col-sum accumulator, then 1/De
    float* w1t   = wsa((size_t)NB * DH);     // W1^T  [NB][DH]
    float* w2t   = wsa((size_t)DH * NC);     // W2^T  [DH][NC]
    float* Pp    = wsa((size_t)Nn * DH);     // d ⊙ (X W1^T + b1)      [Nn][DH]
    float* Ut    = wsa((size_t)DH * Ee);     // (H^T Pp)^T             [DH][Ee]
    float* V     = wsa((size_t)Ee * DH);     // invDe ⊙ U              [Ee][DH]
    float* M1    = wsa((size_t)Nn * DH);     // d ⊙ (H V)  -> h (in place after BN/LReLU)
    float* Qp    = wsa((size_t)Nn * NC);     // d ⊙ (h W2^T + b2)      [Nn][NC]
    float* U2t   = wsa((size_t)NC * Ee);     // (H^T Qp)^T             [NC][Ee]
    float* V2    = wsa((size_t)Ee * NC);     // invDe ⊙ U2             [Ee][NC]
    float* sacc  = wsa(DH);
    float* sqacc = wsa(DH);
    float* bnA   = wsa(DH);
    float* bnB   = wsa(DH);
    (void)n_in; (void)out_size; (void)ws_size;

    auto zero = [&](float* p, size_t n) {
        int g = (int)((n + 255) / 256); if (g > 2048) g = 2048;
        zerof_kernel<<<g, 256, 0, stream>>>(p, (long)n);
    };

    // split-K accumulators (and atomic targets) must start at zero every call
    zero(invDe, Ee);
    zero(Ut,  (size_t)DH * Ee);
    zero(M1,  (size_t)Nn * DH);
    zero(U2t, (size_t)NC * Ee);
    zero(sacc, DH);
    zero(sqacc, DH);
    zero(out, (size_t)Nn * NC);

    // degrees
    rowsum_rsqrt_kernel<<<Nn, 256, 0, stream>>>(H, dvec, Nn, Ee);
    {
        int chunk = (Nn + 63) / 64;
        dim3 g((Ee + 255) / 256, 64);
        colsum_partial_kernel<<<g, 256, 0, stream>>>(H, invDe, Nn, Ee, chunk);
        recip_inplace_kernel<<<(Ee + 255) / 256, 256, 0, stream>>>(invDe, Ee);
    }
    // weight transposes (tiny, L2 resident afterwards)
    {
        dim3 b(32, 8);
        dim3 g1((NB + 31) / 32, (DH + 31) / 32);
        transpose_scaled_kernel<<<g1, b, 0, stream>>>(W1, w1t, DH, NB, nullptr);
        dim3 g2((DH + 31) / 32, (NC + 31) / 32);
        transpose_scaled_kernel<<<g2, b, 0, stream>>>(W2, w2t, NC, DH, nullptr);
    }

    // Pp = d ⊙ (X @ W1^T + b1)
    {
        dim3 g(1, (Nn + 127) / 128, 1);
        gemm_f32_wmma_kernel<true><<<g, 256, 0, stream>>>(
            X, NB, w1t, DH, Pp, DH, Nn, DH, NB, NB, b1, dvec, 0);
    }
    // Ut = (Pp)^T @ H   [DH][Ee], split-K over Nn
    {
        const int S = 8, kc = (Nn + S - 1) / S;
        dim3 g((Ee + 127) / 128, 1, S);
        gemm_f32_wmma_kernel<false><<<g, 256, 0, stream>>>(
            Pp, DH, H, Ee, Ut, Ee, DH, Ee, Nn, kc, nullptr, nullptr, 1);
    }
    // V = invDe ⊙ Ut^T
    {
        dim3 b(32, 8), g((Ee + 31) / 32, (DH + 31) / 32);
        transpose_scaled_kernel<<<g, b, 0, stream>>>(Ut, V, DH, Ee, invDe);
    }
    // M1 = d ⊙ (H @ V)   [Nn][DH], split-K over Ee
    {
        const int S = 4, kc = (Ee + S - 1) / S;
        dim3 g(1, (Nn + 127) / 128, S);
        gemm_f32_wmma_kernel<true><<<g, 256, 0, stream>>>(
            H, Ee, V, DH, M1, DH, Nn, DH, Ee, kc, nullptr, dvec, 1);
    }
    // BatchNorm (training stats) + LeakyReLU(0.1), in place
    {
        const int NBLK = 256, chunk = (Nn + NBLK - 1) / NBLK;
        bn_stats_kernel<<<NBLK, DH, 0, stream>>>(M1, sacc, sqacc, Nn, chunk);
        bn_finalize_kernel<<<1, DH, 0, stream>>>(sacc, sqacc, gamma, beta, bnA, bnB, Nn);
        long tot = (long)Nn * DH;
        int g = (int)((tot + 255) / 256); if (g > 2048) g = 2048;
        bn_apply_lrelu_kernel<<<g, 256, 0, stream>>>(M1, bnA, bnB, tot, DH);
    }
    // Qp = d ⊙ (h @ W2^T + b2)
    {
        dim3 g((NC + 127) / 128, (Nn + 127) / 128, 1);
        gemm_f32_wmma_kernel<true><<<g, 256, 0, stream>>>(
            M1, DH, w2t, NC, Qp, NC, Nn, NC, DH, DH, b2, dvec, 0);
    }
    // U2t = (Qp)^T @ H   [NC][Ee], split-K over Nn
    {
        const int S = 8, kc = (Nn + S - 1) / S;
        dim3 g((Ee + 127) / 128, 1, S);
        gemm_f32_wmma_kernel<false><<<g, 256, 0, stream>>>(
            Qp, NC, H, Ee, U2t, Ee, NC, Ee, Nn, kc, nullptr, nullptr, 1);
    }
    // V2 = invDe ⊙ U2t^T
    {
        dim3 b(32, 8), g((Ee + 31) / 32, (NC + 31) / 32);
        transpose_scaled_kernel<<<g, b, 0, stream>>>(U2t, V2, NC, Ee, invDe);
    }
    // out = d ⊙ (H @ V2)   [Nn][NC], split-K over Ee
    {
        const int S = 4, kc = (Ee + S - 1) / S;
        dim3 g((NC + 127) / 128, (Nn + 127) / 128, S);
        gemm_f32_wmma_kernel<true><<<g, 256, 0, stream>>>(
            H, Ee, V2, NC, out, NC, Nn, NC, Ee, kc, nullptr, dvec, 1);
    }
}